// TemporalCore_46660524704458
// MI455X (gfx1250) — compile-verified
//
#include <hip/hip_runtime.h>
#include <math.h>

#define D      4096
#define D3     12288
#define D6     24576
#define NSTEP  64
#define LN_EPS 1e-5f
#define GBLKS  (D3 / 8)   // 1536 blocks, 8 wave32 rows per block

typedef __attribute__((ext_vector_type(2))) float v2f;
typedef __attribute__((ext_vector_type(8))) float v8f;
typedef __attribute__((ext_vector_type(4))) int   v4i;

// ---------------------------------------------------------------------------
// gfx1250 async global->LDS helpers (guarded; fall back to plain LDS stores).
// Builtin signature (from compiler diagnostic): param0 is global int4*,
// so the call form is (global v4i*, lds v4i*, imm offset, imm cpol).
// ---------------------------------------------------------------------------
__device__ __forceinline__ void async_copy16(const float* __restrict__ g, float* l) {
#if __has_builtin(__builtin_amdgcn_global_load_async_to_lds_b128)
  __builtin_amdgcn_global_load_async_to_lds_b128(
      (__attribute__((address_space(1))) v4i*)g,
      (__attribute__((address_space(3))) v4i*)l,
      /*offset=*/0, /*cpol=*/0);
#else
  *(float4*)l = *(const float4*)g;
#endif
}

__device__ __forceinline__ void async_wait0() {
#if __has_builtin(__builtin_amdgcn_s_wait_asynccnt)
  __builtin_amdgcn_s_wait_asynccnt(0);
#else
  asm volatile("s_wait_asynccnt 0x0" ::: "memory");
#endif
}

// ---------------------------------------------------------------------------
// Build padded ctx_in matrix XIN (D6 x 16 row-major). Columns 0..2 are the
// three distinct ctx_in vectors (step 0, step 1, steps >= 2); cols 3..15 = 0.
// (v = z0 - z_prev collapses to 0 from step 1 on; a collapses to 0 from
//  step 2 on, so ctx/gi take only three distinct values across the scan.)
// ---------------------------------------------------------------------------
__global__ void build_xin_kernel(const float* __restrict__ z0,
                                 const float* __restrict__ mf,
                                 const float* __restrict__ mm,
                                 const float* __restrict__ ms,
                                 const float* __restrict__ zp,
                                 const float* __restrict__ vp,
                                 float* __restrict__ XIN) {
  int j = blockIdx.x * blockDim.x + threadIdx.x;
  if (j >= D6) return;
  int seg = j >> 12;        // j / D
  int r   = j & (D - 1);    // j % D
  float c0 = 0.f, c1 = 0.f, c2 = 0.f;
  if (seg == 0)      { float v = z0[r]; c0 = v; c1 = v; c2 = v; }
  else if (seg == 1) { float v = mf[r]; c0 = v; c1 = v; c2 = v; }
  else if (seg == 2) { float v = mm[r]; c0 = v; c1 = v; c2 = v; }
  else if (seg == 3) { float v = ms[r]; c0 = v; c1 = v; c2 = v; }
  else if (seg == 4) { c0 = z0[r] - zp[r]; }                                  // v
  else               { float v0 = z0[r] - zp[r]; c0 = v0 - vp[r]; c1 = -v0; } // a
  float* row = XIN + (size_t)j * 16;
#pragma unroll
  for (int t = 3; t < 16; ++t) row[t] = 0.f;
  row[0] = c0; row[1] = c1; row[2] = c2;
}

// ---------------------------------------------------------------------------
// WMMA fp32 GEMM with N padded to 16:  OUT(M x 16) = A(M x K) @ B(K x 16).
// One wave per 16-row M tile; grids sized exactly so EXEC is all ones
// during V_WMMA_F32_16X16X4_F32 (ISA requirement).
// mode 0: outP[m*16 + n] = tanh(acc + bias[m])           (padded, for GEMM2)
// mode 1: out3[n*D3 + m] = acc + bias[m]   for n < 3     (the 3 gi variants)
// ---------------------------------------------------------------------------
__global__ void __launch_bounds__(256) wmma_gemm_n16_kernel(
    const float* __restrict__ A, const float* __restrict__ B,
    const float* __restrict__ bias, float* __restrict__ outP,
    float* __restrict__ out3, int K, int mode) {
  int lane = threadIdx.x & 31;
  int wave = blockIdx.x * (blockDim.x >> 5) + (threadIdx.x >> 5);
  int i0   = wave << 4;                 // M-tile base row
  int mrow = lane & 15;                 // also the N index for B/C/D
  int kh   = (lane >> 4) << 1;          // 0 for lanes 0-15, 2 for lanes 16-31

  const float* Ap = A + (size_t)(i0 + mrow) * K + kh;
  const float* Bp = B + (size_t)kh * 16 + mrow;

  v8f acc = {0.f, 0.f, 0.f, 0.f, 0.f, 0.f, 0.f, 0.f};
  for (int j = 0; j < K; j += 4) {
    v2f a = {Ap[0], Ap[1]};
    v2f b = {Bp[0], Bp[16]};
    acc = __builtin_amdgcn_wmma_f32_16x16x4_f32(
        false, a, false, b, (short)0, acc, false, false);
    Ap += 4;
    Bp += 64;   // 4 K-rows * 16 columns
  }

  int mhi = (lane >> 4) << 3;
  int n   = lane & 15;
#pragma unroll
  for (int r = 0; r < 8; ++r) {
    int m   = i0 + r + mhi;
    float u = acc[r] + bias[m];
    if (mode == 0) {
      outP[(size_t)m * 16 + n] = tanhf(u);
    } else if (n < 3) {
      out3[(size_t)n * D3 + m] = u;
    }
  }
}

// ---------------------------------------------------------------------------
// 256-thread (8-wave) block reduction.
// ---------------------------------------------------------------------------
__device__ __forceinline__ float block_reduce_sum256(float v, float* sh) {
#pragma unroll
  for (int off = 16; off > 0; off >>= 1) v += __shfl_down(v, off, 32);
  int wid  = threadIdx.x >> 5;
  int lane = threadIdx.x & 31;
  if (lane == 0) sh[wid] = v;
  __syncthreads();
  if (wid == 0) {
    float t = (lane < 8) ? sh[lane] : 0.f;
#pragma unroll
    for (int off = 4; off > 0; off >>= 1) t += __shfl_down(t, off, 32);
    if (lane == 0) sh[0] = t;
  }
  __syncthreads();
  float r = sh[0];
  __syncthreads();
  return r;
}

// ---------------------------------------------------------------------------
// Fused recurrent step:
//   gh = w_hh @ h + b_hh     (wave-per-row GEMV; h staged in LDS via the
//                             gfx1250 async global->LDS path)
//   then the LAST block to finish (atomic counter) computes the GRU gates,
//   residual and layernorm for the whole vector and writes h_next.
// ---------------------------------------------------------------------------
__global__ void __launch_bounds__(256) gemv_step_kernel(
    const float* __restrict__ W, const float* __restrict__ bias,
    const float* __restrict__ gi, const float* __restrict__ H,
    const float* __restrict__ gamma, const float* __restrict__ beta,
    float* __restrict__ hout, float* __restrict__ gh,
    unsigned int* __restrict__ counter) {
  __shared__ float hs[D];            // 16 KB staged copy of h
  __shared__ float red[8];
  __shared__ unsigned int lastFlag;

  int tid = threadIdx.x;

  // ---- stage h (16 KB) into LDS: 4 x b128 per thread, ASYNCcnt tracked ----
#pragma unroll
  for (int q = 0; q < 4; ++q) {
    int off = (tid + q * 256) * 4;   // float index, 16B granules
    async_copy16(H + off, &hs[off]);
  }
  async_wait0();
  __syncthreads();

  // ---- wave-per-row GEMV out of LDS ----
  int lane = tid & 31;
  int row  = blockIdx.x * 8 + (tid >> 5);
  const float* w = W + (size_t)row * D;
  float s = 0.f;
  for (int j = lane * 4; j < D; j += 128) {
    if (j + 1024 < D) __builtin_prefetch(w + j + 1024, 0, 1);
    float4 wv = *(const float4*)(w + j);
    float4 hv = *(const float4*)(&hs[j]);
    s = fmaf(wv.x, hv.x, s);
    s = fmaf(wv.y, hv.y, s);
    s = fmaf(wv.z, hv.z, s);
    s = fmaf(wv.w, hv.w, s);
  }
#pragma unroll
  for (int off = 16; off > 0; off >>= 1) s += __shfl_down(s, off, 32);
  if (lane == 0) gh[row] = s + bias[row];

  // ---- last arriving block runs the gates + layernorm epilogue ----
  __threadfence();
  __syncthreads();
  if (tid == 0) {
    unsigned int prev = atomicAdd(counter, 1u);
    lastFlag = (prev == (unsigned)(GBLKS - 1)) ? 1u : 0u;
  }
  __syncthreads();
  if (!lastFlag) return;
  __threadfence();   // acquire: all blocks' gh writes visible

  float x[16];
  float ssum = 0.f;
#pragma unroll
  for (int q = 0; q < 16; ++q) {
    int i = tid + q * 256;
    float ir  = gi[i],           hr = gh[i];
    float iz  = gi[D + i],       hz = gh[D + i];
    float in_ = gi[2 * D + i],   hn = gh[2 * D + i];
    float rg = 1.f / (1.f + expf(-(ir + hr)));
    float zg = 1.f / (1.f + expf(-(iz + hz)));
    float ng = tanhf(in_ + rg * hn);
    float hc = hs[i];                       // h from the LDS staging copy
    float xv = hc + (1.f - zg) * ng + zg * hc;   // h + dh
    x[q] = xv;
    ssum += xv;
  }
  float mu = block_reduce_sum256(ssum, red) * (1.f / (float)D);
  float s2 = 0.f;
#pragma unroll
  for (int q = 0; q < 16; ++q) { float dx = x[q] - mu; s2 += dx * dx; }
  float var  = block_reduce_sum256(s2, red) * (1.f / (float)D);
  float rstd = rsqrtf(var + LN_EPS);
#pragma unroll
  for (int q = 0; q < 16; ++q) {
    int i = tid + q * 256;
    hout[i] = (x[q] - mu) * rstd * gamma[i] + beta[i];
  }
  __syncthreads();
  if (tid == 0) *counter = 0u;   // re-arm for the next step's launch
}

// ---------------------------------------------------------------------------
extern "C" void kernel_launch(void* const* d_in, const int* in_sizes, int n_in,
                              void* d_out, int out_size, void* d_ws, size_t ws_size,
                              hipStream_t stream) {
  const float* h0    = (const float*)d_in[0];
  const float* z0    = (const float*)d_in[1];
  const float* W_ctx = (const float*)d_in[2];
  const float* b_ctx = (const float*)d_in[3];
  const float* w_ih  = (const float*)d_in[4];
  const float* w_hh  = (const float*)d_in[5];
  const float* b_ih  = (const float*)d_in[6];
  const float* b_hh  = (const float*)d_in[7];
  const float* gamma = (const float*)d_in[8];
  const float* beta  = (const float*)d_in[9];
  const float* mf    = (const float*)d_in[10];
  const float* mm    = (const float*)d_in[11];
  const float* ms    = (const float*)d_in[12];
  const float* zp    = (const float*)d_in[13];
  const float* vp    = (const float*)d_in[14];
  float* out = (float*)d_out;

  float* ws   = (float*)d_ws;
  float* XIN  = ws;                         // D6*16 floats  (1.5 MB)
  float* CTXP = XIN + (size_t)D6 * 16;      // D*16 floats
  float* GI3  = CTXP + (size_t)D * 16;      // 3*D3 floats
  float* GH   = GI3 + (size_t)3 * D3;       // D3 floats
  float* H    = GH + D3;                    // D floats
  unsigned int* counter = (unsigned int*)(H + D);

  (void)hipMemcpyAsync(H, h0, D * sizeof(float), hipMemcpyDeviceToDevice, stream);
  (void)hipMemsetAsync(counter, 0, sizeof(unsigned int), stream);

  build_xin_kernel<<<(D6 + 255) / 256, 256, 0, stream>>>(z0, mf, mm, ms, zp, vp, XIN);

  // ctx variants: CTXP = tanh(W_ctx @ XIN + b_ctx)   (4096 x 16, WMMA)
  wmma_gemm_n16_kernel<<<(D / 16) / 8, 256, 0, stream>>>(
      W_ctx, XIN, b_ctx, CTXP, nullptr, D6, 0);
  // gi variants: GI3[v] = w_ih @ ctx_v + b_ih        (3 x 12288, WMMA)
  wmma_gemm_n16_kernel<<<(D3 / 16) / 8, 256, 0, stream>>>(
      w_ih, CTXP, b_ih, nullptr, GI3, D, 1);

  for (int t = 0; t < NSTEP; ++t) {
    const float* gi = GI3 + (size_t)(t < 2 ? t : 2) * D3;
    float* hout = (t == NSTEP - 1) ? out : H;
    gemv_step_kernel<<<GBLKS, 256, 0, stream>>>(
        w_hh, b_hh, gi, H, gamma, beta, hout, GH, counter);
  }
}